// YOLOv8Loss_21449066676695
// MI455X (gfx1250) — compile-verified
//
#include <hip/hip_runtime.h>
#include <hip/hip_bf16.h>
#include <math.h>

// ---------------------------------------------------------------------------
// YOLOv8 loss, fused for MI455X (gfx1250, wave32).
//   B=32, G=32, N=8400 anchors (80x80 s8, 40x40 s16, 20x20 s32), 144 ch.
// Bandwidth-bound: the only dense pass (k_decode_bce) streams p0..p2 once.
// Wave reductions use V_WMMA_F32_16X16X4_F32 with an all-ones B matrix:
//   D[m,n] = x(lane m) + x(lane m+16)  ->  per-lane sum of 8 acc VGPRs +
//   one shfl_xor(16) == exact f32 wave32 sum in a single matrix op.
// ---------------------------------------------------------------------------

#define NB     32
#define NG     32
#define NANCH  8400
#define NCH    144
#define NCLS   80
#define MASKW  264          // ceil(8400/32)=263, rounded to 264 (8-word chunks)
#define EPSF   1e-9f

typedef float v2f __attribute__((ext_vector_type(2)));
typedef float v8f __attribute__((ext_vector_type(8)));

// Workspace layout (bytes). Total ~11.85 MB (fits easily in 192MB L2).
#define PRED_OFF 0u           // float4[B*N]      pred boxes
#define LSE_OFF  4300800u     // float4[B*N]      per-side DFL logsumexp
#define ALM_OFF  8601600u     // float [B*N]      align value at matched gt
#define MAT_OFF  9676800u     // int   [B*N]      matched gt (-1 = background)
#define AMAX_OFF 10752000u    // uint  [B*G]      pos_align_max (float bits)
#define IMAX_OFF 10756096u    // uint  [B*G]      pos_iou_max   (float bits)
#define MASK_OFF 10760192u    // uint  [B*G*264]  topk bitmask
#define ACC_OFF  11841536u    // float [8]        accumulators

// acc[0]=bce_base  acc[1]=sum s*norm  acc[2]=sum(1-ciou)  acc[3]=dfl  acc[4]=num_pos

// --------------------------- helpers ---------------------------------------

__device__ __forceinline__ void anchor_info(int n, int& lvl, int& plane, int& hw,
                                            float& stride, float& cx, float& cy) {
  int Wd;
  if (n < 6400)      { lvl = 0; plane = 6400; hw = n;        Wd = 80; stride = 8.f;  }
  else if (n < 8000) { lvl = 1; plane = 1600; hw = n - 6400; Wd = 40; stride = 16.f; }
  else               { lvl = 2; plane = 400;  hw = n - 8000; Wd = 20; stride = 32.f; }
  int iy = hw / Wd, ix = hw - iy * Wd;
  cx = (ix + 0.5f) * stride;
  cy = (iy + 0.5f) * stride;
}

__device__ __forceinline__ const float* lvl_base(const float* p0, const float* p1,
                                                 const float* p2, int lvl, int b,
                                                 int plane, int hw) {
  const float* P = (lvl == 0) ? p0 : ((lvl == 1) ? p1 : p2);
  return P + (size_t)b * NCH * plane + hw;
}

// Exact f32 wave32 sum via one V_WMMA_F32_16X16X4_F32 (B = ones => layout-free).
// EXEC must be all ones at call sites (all grids are exact multiples of 256).
__device__ __forceinline__ float wave_sum_wmma(float x) {
  v2f a; a[0] = x;   a[1] = 0.0f;   // A: 16x4, K0/K2 carry lane data, K1/K3 zero
  v2f bm; bm[0] = 1.0f; bm[1] = 1.0f; // B: 4x16 all ones
  v8f c = {};
  c = __builtin_amdgcn_wmma_f32_16x16x4_f32(false, a, false, bm,
                                            (short)0, c, false, false);
  // lane L (L<16):  c[r] = D[r,   L] = sum_{m=r}       rows 0..7
  // lane L (L>=16): c[r] = D[8+r, L-16]                rows 8..15
  float s = c[0] + c[1] + c[2] + c[3] + c[4] + c[5] + c[6] + c[7];
  s += __shfl_xor(s, 16, 32);   // rows 0..7 partial + rows 8..15 partial
  return s;                     // every lane: total over 32 lanes
}

// --------------------------- kernels ---------------------------------------

__global__ void k_init(unsigned* amax, unsigned* imax, float* acc) {
  int t = blockIdx.x * 256 + threadIdx.x;
  if (t < NB * NG) { amax[t] = 0u; imax[t] = 0u; }
  if (t < 8) acc[t] = 0.f;
}

// Dense pass: DFL decode (+logsumexp) and BCE base term. Streams p* once.
__global__ void __launch_bounds__(256) k_decode_bce(
    const float* __restrict__ p0, const float* __restrict__ p1,
    const float* __restrict__ p2, float4* __restrict__ pred,
    float4* __restrict__ lse_out, float* __restrict__ acc) {
  int idx = blockIdx.x * 256 + threadIdx.x;          // grid is exactly B*N/256
  int b = idx / NANCH, n = idx - b * NANCH;
  int lvl, plane, hw; float stride, cx, cy;
  anchor_info(n, lvl, plane, hw, stride, cx, cy);
  const float* base = lvl_base(p0, p1, p2, lvl, b, plane, hw);

  float dist[4], lse[4];
#pragma unroll
  for (int s4 = 0; s4 < 4; ++s4) {
    float x[16];
#pragma unroll
    for (int j = 0; j < 16; ++j) x[j] = base[(s4 * 16 + j) * plane];
    float m = x[0];
#pragma unroll
    for (int j = 1; j < 16; ++j) m = fmaxf(m, x[j]);
    float se = 0.f, ej = 0.f;
#pragma unroll
    for (int j = 0; j < 16; ++j) {
      float e = expf(x[j] - m);
      se += e;
      ej += e * (float)j;
    }
    lse[s4]  = m + logf(se);
    dist[s4] = (ej / se) * stride;
  }
  pred[idx]    = make_float4(cx - dist[0], cy - dist[1], cx + dist[2], cy + dist[3]);
  lse_out[idx] = make_float4(lse[0], lse[1], lse[2], lse[3]);

  // BCE base: sum over classes of max(s,0)+log1p(exp(-|s|))
  __builtin_prefetch(base + 64 * plane, 0, 1);       // global_prefetch_b8
  float accv = 0.f;
#pragma unroll 4
  for (int c = 0; c < NCLS; ++c) {
    float s = base[(64 + c) * plane];
    accv += fmaxf(s, 0.f) + log1pf(expf(-fabsf(s)));
  }
  float ws = wave_sum_wmma(accv);
  if ((threadIdx.x & 31) == 0) atomicAdd(&acc[0], ws);
}

// One workgroup per (b,g): align metric over 8400 anchors in LDS, then 10
// masked-argmax rounds -> packed top-k bitmask.
__global__ void __launch_bounds__(256) k_topk(
    const float* __restrict__ p0, const float* __restrict__ p1,
    const float* __restrict__ p2, const float4* __restrict__ pred,
    const float4* __restrict__ gt, const int* __restrict__ labels,
    unsigned* __restrict__ maskws) {
  __shared__ float    s_align[NANCH];   // 33.6 KB of the 320 KB WGP LDS
  __shared__ unsigned s_mask[MASKW];
  __shared__ float    s_rv[256];
  __shared__ int      s_ri[256];

  int bg = blockIdx.x;
  int b  = bg >> 5;
  int tid = threadIdx.x;
  float4 gt4 = gt[bg];
  int   lab  = labels[bg];
  bool  mgt  = (gt4.x + gt4.y + gt4.z + gt4.w) > 0.f;
  float ag   = fmaxf(gt4.z - gt4.x, 0.f) * fmaxf(gt4.w - gt4.y, 0.f);

  for (int w = tid; w < MASKW; w += 256) s_mask[w] = 0u;

  for (int n = tid; n < NANCH; n += 256) {
    int lvl, plane, hw; float stride, cx, cy;
    anchor_info(n, lvl, plane, hw, stride, cx, cy);
    float4 P = pred[b * NANCH + n];
    float iw = fmaxf(fminf(gt4.z, P.z) - fmaxf(gt4.x, P.x), 0.f);
    float ih = fmaxf(fminf(gt4.w, P.w) - fmaxf(gt4.y, P.y), 0.f);
    float inter = iw * ih;
    float ap  = fmaxf(P.z - P.x, 0.f) * fmaxf(P.w - P.y, 0.f);
    float iou = inter / (ag + ap - inter + 1e-7f);
    bool ing = (cx > gt4.x) && (cx < gt4.z) && (cy > gt4.y) && (cy < gt4.w);
    const float* base = lvl_base(p0, p1, p2, lvl, b, plane, hw);
    float sc = base[(64 + lab) * plane];
    float cg = 1.f / (1.f + expf(-sc));
    float i2 = iou * iou;
    s_align[n] = (ing && mgt) ? sqrtf(cg) * i2 * i2 * i2 : 0.f;
  }
  __syncthreads();

  for (int k = 0; k < 10; ++k) {
    float bv = -1.f; int bi = -1;
    for (int n = tid; n < NANCH; n += 256) {
      float v = s_align[n];
      if (v > bv) { bv = v; bi = n; }
    }
    s_rv[tid] = bv; s_ri[tid] = bi;
    __syncthreads();
    for (int off = 128; off; off >>= 1) {
      if (tid < off) {
        float v2 = s_rv[tid + off]; int i2 = s_ri[tid + off];
        if (v2 > s_rv[tid] || (v2 == s_rv[tid] && i2 >= 0 && i2 < s_ri[tid])) {
          s_rv[tid] = v2; s_ri[tid] = i2;
        }
      }
      __syncthreads();
    }
    if (tid == 0) {
      int i = s_ri[0];
      if (i >= 0) {
        if (s_rv[0] > EPSF) s_mask[i >> 5] |= (1u << (i & 31));
        s_align[i] = -2.f;                // remove from further rounds
      }
    }
    __syncthreads();
  }

  for (int w = tid; w < MASKW; w += 256) maskws[bg * MASKW + w] = s_mask[w];
}

// Per-anchor resolution: count gt matches, break multi-matches by best IoU,
// atomic-max the per-gt align/iou maxima, count positives.
__global__ void __launch_bounds__(256) k_assign(
    const float* __restrict__ p0, const float* __restrict__ p1,
    const float* __restrict__ p2, const float4* __restrict__ pred,
    const float4* __restrict__ gt, const int* __restrict__ labels,
    const unsigned* __restrict__ maskws, int* __restrict__ matched_out,
    float* __restrict__ alm_out, unsigned* __restrict__ amax,
    unsigned* __restrict__ imax, float* __restrict__ acc) {
  __shared__ float4   s_gt[NG];
  __shared__ unsigned s_w[256];            // 32 gts x 8 words (256 anchors)

  int blk = blockIdx.x;
  int b = blk / 33, chunk = blk - b * 33;
  int tid = threadIdx.x;
  int n = chunk * 256 + tid;

  if (tid < NG) s_gt[tid] = gt[b * NG + tid];
  {
    int g = tid >> 3, wo = tid & 7;
    s_w[tid] = maskws[(b * NG + g) * MASKW + chunk * 8 + wo];
  }
  __syncthreads();

  bool active = n < NANCH;
  bool fg = false;
  if (active) {
    int lvl, plane, hw; float stride, cx, cy;
    anchor_info(n, lvl, plane, hw, stride, cx, cy);
    float4 P = pred[b * NANCH + n];
    float ap = fmaxf(P.z - P.x, 0.f) * fmaxf(P.w - P.y, 0.f);

    int count = 0, firstg = -1, bestg = 0;
    float besti = -1.f;
    int wsub = tid >> 5;
    unsigned bit = 1u << (tid & 31);
    for (int g = 0; g < NG; ++g) {
      float4 T = s_gt[g];
      float iw = fmaxf(fminf(T.z, P.z) - fmaxf(T.x, P.x), 0.f);
      float ih = fmaxf(fminf(T.w, P.w) - fmaxf(T.y, P.y), 0.f);
      float inter = iw * ih;
      float agv = fmaxf(T.z - T.x, 0.f) * fmaxf(T.w - T.y, 0.f);
      float iou = inter / (agv + ap - inter + 1e-7f);
      if (iou > besti) { besti = iou; bestg = g; }
      if (s_w[g * 8 + wsub] & bit) { if (count == 0) firstg = g; ++count; }
    }

    int matched = -1;
    float alm = 0.f;
    if (count > 0) {
      fg = true;
      matched = (count > 1) ? bestg : firstg;
      float4 T = s_gt[matched];
      float iw = fmaxf(fminf(T.z, P.z) - fmaxf(T.x, P.x), 0.f);
      float ih = fmaxf(fminf(T.w, P.w) - fmaxf(T.y, P.y), 0.f);
      float inter = iw * ih;
      float agv = fmaxf(T.z - T.x, 0.f) * fmaxf(T.w - T.y, 0.f);
      float iou = inter / (agv + ap - inter + 1e-7f);
      bool ing = (cx > T.x) && (cx < T.z) && (cy > T.y) && (cy < T.w);
      bool mgt = (T.x + T.y + T.z + T.w) > 0.f;
      const float* base = lvl_base(p0, p1, p2, lvl, b, plane, hw);
      int lab = labels[b * NG + matched];
      float sc = base[(64 + lab) * plane];
      float cg = 1.f / (1.f + expf(-sc));
      float i2 = iou * iou;
      alm = (ing && mgt) ? sqrtf(cg) * i2 * i2 * i2 : 0.f;
      atomicMax(&amax[b * NG + matched], __float_as_uint(alm));
      atomicMax(&imax[b * NG + matched], __float_as_uint(iou));
    }
    matched_out[b * NANCH + n] = matched;
    alm_out[b * NANCH + n]     = alm;
  }

  unsigned long long bal = __ballot(active && fg);
  if ((tid & 31) == 0) atomicAdd(&acc[4], (float)__popcll(bal));
}

// Positive-anchor losses: cls target term, CIoU, DFL (via stored logsumexp).
__global__ void __launch_bounds__(256) k_loss(
    const float* __restrict__ p0, const float* __restrict__ p1,
    const float* __restrict__ p2, const float4* __restrict__ pred,
    const float4* __restrict__ lse4, const float4* __restrict__ gt,
    const int* __restrict__ labels, const int* __restrict__ matched_in,
    const float* __restrict__ alm_in, const unsigned* __restrict__ amax,
    const unsigned* __restrict__ imax, float* __restrict__ acc) {
  int idx = blockIdx.x * 256 + threadIdx.x;          // exactly B*N threads
  int b = idx / NANCH, n = idx - b * NANCH;
  int matched = matched_in[idx];

  float cls_pos = 0.f, iou_l = 0.f, dfl_l = 0.f;
  if (matched >= 0) {
    int lvl, plane, hw; float stride, cx, cy;
    anchor_info(n, lvl, plane, hw, stride, cx, cy);
    const float* base = lvl_base(p0, p1, p2, lvl, b, plane, hw);
    float4 P = pred[idx];
    float4 T = gt[b * NG + matched];

    float am = __uint_as_float(amax[b * NG + matched]);
    float im = __uint_as_float(imax[b * NG + matched]);
    float norm = alm_in[idx] * im / (am + EPSF);

    int lab = labels[b * NG + matched];
    cls_pos = base[(64 + lab) * plane] * norm;       // the  s*t  BCE term

    // ---- CIoU (matches reference ciou(), eps=1e-7) ----
    float w1 = P.z - P.x, h1 = P.w - P.y;
    float w2 = T.z - T.x, h2 = T.w - T.y;
    float iw = fmaxf(fminf(P.z, T.z) - fmaxf(P.x, T.x), 0.f);
    float ih = fmaxf(fminf(P.w, T.w) - fmaxf(P.y, T.y), 0.f);
    float inter = iw * ih;
    float uni = w1 * h1 + w2 * h2 - inter + 1e-7f;
    float iou = inter / uni;
    float cw = fmaxf(P.z, T.z) - fminf(P.x, T.x);
    float ch = fmaxf(P.w, T.w) - fminf(P.y, T.y);
    float c2 = cw * cw + ch * ch + 1e-7f;
    float dx = P.x + P.z - T.x - T.z;
    float dy = P.y + P.w - T.y - T.w;
    float rho2 = (dx * dx + dy * dy) * 0.25f;
    float dv = atanf(w2 / (h2 + 1e-7f)) - atanf(w1 / (h1 + 1e-7f));
    float v  = 0.40528473456935109f * dv * dv;       // 4/pi^2
    float al = v / (v - iou + 1.0f + 1e-7f);
    iou_l = 1.f - (iou - rho2 / c2 - al * v);

    // ---- DFL: -(td . log_softmax) = lse - ((1-a)x[lo] + a x[hi]) per side ----
    float4 L = lse4[idx];
    float d[4]   = { fmaxf(cx - T.x, 0.f), fmaxf(cy - T.y, 0.f),
                     fmaxf(T.z - cx, 0.f), fmaxf(T.w - cy, 0.f) };
    float lseA[4] = { L.x, L.y, L.z, L.w };
#pragma unroll
    for (int s4 = 0; s4 < 4; ++s4) {
      float tb = fminf(d[s4], 15.f - 1e-6f);
      float lo = floorf(tb);
      float a  = tb - lo;
      int li = (int)lo;
      int ui = min(li + 1, 15);
      float xlo = base[(s4 * 16 + li) * plane];
      float xhi = base[(s4 * 16 + ui) * plane];
      dfl_l += lseA[s4] - ((1.f - a) * xlo + a * xhi);
    }
  }

  float r1 = wave_sum_wmma(cls_pos);
  float r2 = wave_sum_wmma(iou_l);
  float r3 = wave_sum_wmma(dfl_l);
  if ((threadIdx.x & 31) == 0) {
    atomicAdd(&acc[1], r1);
    atomicAdd(&acc[2], r2);
    atomicAdd(&acc[3], r3);
  }
}

__global__ void k_final(const float* __restrict__ acc, float* __restrict__ out) {
  if (threadIdx.x == 0) {
    float np = fmaxf(acc[4], 1.f);
    out[0] = 7.5f * acc[2] / np;                 // W_IOU * loss_iou
    out[1] = 0.5f * (acc[0] - acc[1]) / np;      // W_CLS * loss_cls
    out[2] = 1.5f * acc[3] / np;                 // W_DFL * loss_dfl
  }
}

// --------------------------- launcher --------------------------------------

extern "C" void kernel_launch(void* const* d_in, const int* in_sizes, int n_in,
                              void* d_out, int out_size, void* d_ws, size_t ws_size,
                              hipStream_t stream) {
  (void)in_sizes; (void)n_in; (void)out_size; (void)ws_size;
  const float* p0  = (const float*)d_in[0];
  const float* p1  = (const float*)d_in[1];
  const float* p2  = (const float*)d_in[2];
  const float4* gt = (const float4*)d_in[3];
  const int* lab   = (const int*)d_in[4];

  char* ws = (char*)d_ws;
  float4*   pred = (float4*)(ws + PRED_OFF);
  float4*   lse  = (float4*)(ws + LSE_OFF);
  float*    alm  = (float*)(ws + ALM_OFF);
  int*      mat  = (int*)(ws + MAT_OFF);
  unsigned* amax = (unsigned*)(ws + AMAX_OFF);
  unsigned* imax = (unsigned*)(ws + IMAX_OFF);
  unsigned* mask = (unsigned*)(ws + MASK_OFF);
  float*    acc  = (float*)(ws + ACC_OFF);
  float*    out  = (float*)d_out;

  const int total_blocks = (NB * NANCH) / 256;   // 1050, exact (EXEC-safe WMMA)

  k_init      <<<4, 256, 0, stream>>>(amax, imax, acc);
  k_decode_bce<<<total_blocks, 256, 0, stream>>>(p0, p1, p2, pred, lse, acc);
  k_topk      <<<NB * NG, 256, 0, stream>>>(p0, p1, p2, pred, gt, lab, mask);
  k_assign    <<<NB * 33, 256, 0, stream>>>(p0, p1, p2, pred, gt, lab, mask,
                                            mat, alm, amax, imax, acc);
  k_loss      <<<total_blocks, 256, 0, stream>>>(p0, p1, p2, pred, lse, gt, lab,
                                                 mat, alm, amax, imax, acc);
  k_final     <<<1, 32, 0, stream>>>(acc, out);
}